// MSE_SSIMLoss_9045201125741
// MI455X (gfx1250) — compile-verified
//
#include <hip/hip_runtime.h>

// ---------------------------------------------------------------------------
// MSE + SSIM loss, fused, using V_WMMA_F32_16X16X4_F32 for both 1D Gaussian
// conv passes (separable 11x11 window), f32 end to end.
// ---------------------------------------------------------------------------

typedef float v2f __attribute__((ext_vector_type(2)));
typedef float v8f __attribute__((ext_vector_type(8)));

#define IMG        512
#define TILE_X     16
#define TILE_Y     64
#define HALO       5
#define RROWS      80    // 74 valid halo rows, padded to 5 x 16
#define RCOLS      28    // 26 valid halo cols, padded even (8B-aligned b64 A loads)
#define MIDSTRIDE  17    // bank-conflict-free mid stride
#define NPLANES    96    // 32 batch * 3 channels
#define TPP        256   // tiles per 512x512 plane (32 x * 8 y)
#define NWG        (NPLANES * TPP)
#define C1_CONST   1.0e-4f
#define C2_CONST   9.0e-4f

// Normalized 11-tap Gaussian (sigma=1.5), selected by compare chain so the
// per-lane banded-matrix constants stay in VGPRs (no scratch indexing).
__device__ __forceinline__ float wtap(int idx) {
  float v = 0.0f;
  v = (idx == 0)  ? 0.00102838f : v;
  v = (idx == 1)  ? 0.00759876f : v;
  v = (idx == 2)  ? 0.03600077f : v;
  v = (idx == 3)  ? 0.10936116f : v;
  v = (idx == 4)  ? 0.21300553f : v;
  v = (idx == 5)  ? 0.26601172f : v;
  v = (idx == 6)  ? 0.21300553f : v;
  v = (idx == 7)  ? 0.10936116f : v;
  v = (idx == 8)  ? 0.03600077f : v;
  v = (idx == 9)  ? 0.00759876f : v;
  v = (idx == 10) ? 0.00102838f : v;
  return v;
}

__device__ __forceinline__ v8f wmma4(v2f a, v2f b, v8f c) {
  return __builtin_amdgcn_wmma_f32_16x16x4_f32(false, a, false, b, (short)0, c,
                                               false, false);
}

__global__ __launch_bounds__(128) void ssim_mse_tile_kernel(
    const float* __restrict__ pred, const float* __restrict__ targ,
    float* __restrict__ partials) {
  __shared__ float sP[RROWS * RCOLS];
  __shared__ float sT[RROWS * RCOLS];
  __shared__ float sMid[5][RROWS * MIDSTRIDE];
  __shared__ float redM[128];
  __shared__ float redS[128];

  const int tid   = threadIdx.x;
  const int lane  = tid & 31;
  const int wave  = tid >> 5;
  const int n     = lane & 15;              // M (A) / N (B,D) index
  const int cb    = ((lane >> 4) & 1) * 2;  // K base for this half-wave
  const int half8 = ((lane >> 4) & 1) * 8;  // D-row offset for this half-wave

  const int wg    = blockIdx.x;
  const int plane = wg >> 8;
  const int tidx  = wg & 255;
  const int x0    = (tidx & 31) * TILE_X;
  const int y0    = (tidx >> 5) * TILE_Y;
  const size_t planeBase = (size_t)plane * (IMG * IMG);

  // ---- phase 0: zero-padded halo load to LDS --------------------------------
  for (int idx = tid; idx < RROWS * RCOLS; idx += 128) {
    int rr = idx / RCOLS;
    int cc = idx - rr * RCOLS;
    int gy = y0 - HALO + rr;
    int gx = x0 - HALO + cc;
    bool ok = (rr < 74) && (cc < 26) && (gy >= 0) && (gy < IMG) && (gx >= 0) &&
              (gx < IMG);
    float pv = 0.0f, tv = 0.0f;
    if (ok) {
      size_t g = planeBase + (size_t)gy * IMG + (size_t)gx;
      pv = pred[g];
      tv = targ[g];
    }
    sP[idx] = pv;
    sT[idx] = tv;
  }
  __syncthreads();

  // MSE over the non-overlapping 16x64 core
  float mseAcc = 0.0f;
  for (int idx = tid; idx < TILE_X * TILE_Y; idx += 128) {
    int rr = (idx >> 4) + HALO;
    int cc = (idx & 15) + HALO;
    float d = sP[rr * RCOLS + cc] - sT[rr * RCOLS + cc];
    mseAcc += d * d;
  }

  // Per-lane banded weight constants: w[4c + cb + i - n] (shared by the
  // horizontal-pass B operand and the vertical-pass A operand).
  float wsel0[7], wsel1[7];
#pragma unroll
  for (int c = 0; c < 7; ++c) {
    wsel0[c] = wtap(4 * c + cb + 0 - n);
    wsel1[c] = wtap(4 * c + cb + 1 - n);
  }

  // ---- phase 1: horizontal Gaussian pass (banded GEMM, K=26 -> 7 WMMAs) -----
  // Mid(16r+m, x) = sum_k P(16r+m, k) * W[k - x];  A = image rows, B = band.
  for (int r = wave; r < 5; r += 4) {
    v8f aP = {}, aT = {}, aPP = {}, aTT = {}, aPT = {};
    const int rowBase = (16 * r + n) * RCOLS;
#pragma unroll
    for (int c = 0; c < 7; ++c) {
      const int col = 4 * c + cb;
      float p0 = sP[rowBase + col];
      float p1 = sP[rowBase + col + 1];
      float t0 = sT[rowBase + col];
      float t1 = sT[rowBase + col + 1];
      v2f B;
      B[0] = wsel0[c];
      B[1] = wsel1[c];
      v2f A;
      A[0] = p0;      A[1] = p1;      aP  = wmma4(A, B, aP);
      A[0] = t0;      A[1] = t1;      aT  = wmma4(A, B, aT);
      A[0] = p0 * p0; A[1] = p1 * p1; aPP = wmma4(A, B, aPP);
      A[0] = t0 * t0; A[1] = t1 * t1; aTT = wmma4(A, B, aTT);
      A[0] = p0 * t0; A[1] = p1 * t1; aPT = wmma4(A, B, aPT);
    }
#pragma unroll
    for (int v = 0; v < 8; ++v) {
      int m = (16 * r + v + half8) * MIDSTRIDE + n;
      sMid[0][m] = aP[v];
      sMid[1][m] = aT[v];
      sMid[2][m] = aPP[v];
      sMid[3][m] = aTT[v];
      sMid[4][m] = aPT[v];
    }
  }
  __syncthreads();

  // ---- phase 2: vertical Gaussian pass; wave w owns output row-tile w -------
  // Out(16w+m, x) = sum_k W[k - m] * Mid(16w + k, x);  A = band, B = mid rows.
  v8f m1 = {}, m2 = {}, qPP = {}, qTT = {}, qPT = {};
  {
    const int tbase = 16 * wave;
#pragma unroll
    for (int c = 0; c < 7; ++c) {
      const int k0 = (tbase + 4 * c + cb) * MIDSTRIDE + n;
      const int k1 = k0 + MIDSTRIDE;
      v2f A;
      A[0] = wsel0[c];
      A[1] = wsel1[c];
      v2f B;
      B[0] = sMid[0][k0]; B[1] = sMid[0][k1]; m1  = wmma4(A, B, m1);
      B[0] = sMid[1][k0]; B[1] = sMid[1][k1]; m2  = wmma4(A, B, m2);
      B[0] = sMid[2][k0]; B[1] = sMid[2][k1]; qPP = wmma4(A, B, qPP);
      B[0] = sMid[3][k0]; B[1] = sMid[3][k1]; qTT = wmma4(A, B, qTT);
      B[0] = sMid[4][k0]; B[1] = sMid[4][k1]; qPT = wmma4(A, B, qPT);
    }
  }

  // ---- SSIM map + per-thread accumulation (layout-agnostic elementwise) -----
  float ssimAcc = 0.0f;
#pragma unroll
  for (int v = 0; v < 8; ++v) {
    float mu1 = m1[v], mu2 = m2[v];
    float mu1s = mu1 * mu1;
    float mu2s = mu2 * mu2;
    float mu12 = mu1 * mu2;
    float s1  = qPP[v] - mu1s;
    float s2  = qTT[v] - mu2s;
    float s12 = qPT[v] - mu12;
    float num = (2.0f * mu12 + C1_CONST) * (2.0f * s12 + C2_CONST);
    float den = (mu1s + mu2s + C1_CONST) * (s1 + s2 + C2_CONST);
    ssimAcc += num / den;
  }

  // ---- block reduction, one deterministic partial pair per workgroup --------
  redM[tid] = mseAcc;
  redS[tid] = ssimAcc;
  __syncthreads();
#pragma unroll
  for (int s = 64; s > 0; s >>= 1) {
    if (tid < s) {
      redM[tid] += redM[tid + s];
      redS[tid] += redS[tid + s];
    }
    __syncthreads();
  }
  if (tid == 0) {
    partials[2 * wg + 0] = redM[0];
    partials[2 * wg + 1] = redS[0];
  }
}

__global__ __launch_bounds__(256) void finalize_kernel(
    const float* __restrict__ partials, float* __restrict__ out) {
  __shared__ float rm[256];
  __shared__ float rs[256];
  float m = 0.0f, s = 0.0f;
  for (int i = threadIdx.x; i < NWG; i += 256) {
    m += partials[2 * i + 0];
    s += partials[2 * i + 1];
  }
  rm[threadIdx.x] = m;
  rs[threadIdx.x] = s;
  __syncthreads();
#pragma unroll
  for (int h = 128; h > 0; h >>= 1) {
    if (threadIdx.x < h) {
      rm[threadIdx.x] += rm[threadIdx.x + h];
      rs[threadIdx.x] += rs[threadIdx.x + h];
    }
    __syncthreads();
  }
  if (threadIdx.x == 0) {
    const float invN = 1.0f / 25165824.0f;  // 32*3*512*512
    float mse  = rm[0] * invN;
    float ssim = rs[0] * invN;
    out[0] = mse + 0.01f * (1.0f - ssim);
  }
}

extern "C" void kernel_launch(void* const* d_in, const int* in_sizes, int n_in,
                              void* d_out, int out_size, void* d_ws,
                              size_t ws_size, hipStream_t stream) {
  (void)in_sizes; (void)n_in; (void)out_size; (void)ws_size;
  const float* pred = (const float*)d_in[0];
  const float* targ = (const float*)d_in[1];
  float* out = (float*)d_out;
  float* ws  = (float*)d_ws;  // needs NWG*2*4 = 196,608 bytes

  ssim_mse_tile_kernel<<<NWG, 128, 0, stream>>>(pred, targ, ws);
  finalize_kernel<<<1, 256, 0, stream>>>(ws, out);
}